// CenterLoss_55886114455908
// MI455X (gfx1250) — compile-verified
//
#include <hip/hip_runtime.h>
#include <hip/hip_bf16.h>
#include <float.h>

typedef __attribute__((ext_vector_type(2))) float v2f;
typedef __attribute__((ext_vector_type(8))) float v8f;

#define FEAT_DIM 96
#define CLIP_MIN 1e-12f
#define CLIP_MAX 1e12f

// ---------------------------------------------------------------------------
// Kernel 1: row-wise argmax over logits [N, C] -> labels[N]
// One 256-thread block per row; memory-bound (135 MB of logits dominates the
// whole problem).  First-max tie-breaking matches jnp.argmax.
// ---------------------------------------------------------------------------
__global__ void argmax_rows_kernel(const float* __restrict__ logits,
                                   int* __restrict__ labels,
                                   int C) {
    const int row = blockIdx.x;
    const int tid = threadIdx.x;
    const float* __restrict__ p = logits + (size_t)row * C;

    float best = -__builtin_inff();
    int bestIdx = 0;
    for (int j = tid; j < C; j += 256) {
        float v = p[j];
        if (v > best) { best = v; bestIdx = j; }   // strict > keeps first max
    }

    __shared__ float svals[256];
    __shared__ int   sidx[256];
    svals[tid] = best;
    sidx[tid]  = bestIdx;
    __syncthreads();
    for (int s = 128; s > 0; s >>= 1) {
        if (tid < s) {
            float v2 = svals[tid + s];
            int   i2 = sidx[tid + s];
            float v1 = svals[tid];
            int   i1 = sidx[tid];
            if (v2 > v1 || (v2 == v1 && i2 < i1)) {
                svals[tid] = v2;
                sidx[tid]  = i2;
            }
        }
        __syncthreads();
    }
    if (tid == 0) labels[row] = sidx[0];
}

// ---------------------------------------------------------------------------
// Kernel 2: per-sample squared distance to own-label center via WMMA.
// Each wave32 handles a tile of 16 samples.  A = feats tile (16 x 96),
// B = gathered centers^T (96 x 16); 24 chained V_WMMA_F32_16X16X4_F32 steps
// accumulate F * G^T; diagonal (i,i) = f_i . c_i.
//
// f32 16x4 A layout (ISA 7.12.2): lane<16 holds (M=lane, K=v), lane>=16 holds
// (M=lane-16, K=2+v).  B (4x16) mirrors it: lane half h, VGPR v -> row K=2h+v,
// col N=lane%16.  So each lane loads 2 consecutive K-elements of its row for
// both A and B: base K = 4*step + 2*(lane/16).
// Diagonal extraction: lane i (i<8) holds D[i][i] in c[i]; lane i+16 (i>=8)
// holds D[i][i] in c[i-8]; both also hold row i's sum-of-squares after a
// shfl_xor(...,16) combine.
// ---------------------------------------------------------------------------
__global__ void center_dist_wmma_kernel(const float* __restrict__ feats,
                                        const float* __restrict__ centers,
                                        const int* __restrict__ labels,
                                        float* __restrict__ partials,
                                        int ntiles) {
    const int lane   = threadIdx.x & 31;
    const int waveId = threadIdx.x >> 5;
    const int tile   = blockIdx.x * 8 + waveId;
    if (tile >= ntiles) return;                    // wave-uniform: EXEC stays full

    const int r    = lane & 15;                    // sample row within tile
    const int half = lane >> 4;                    // K-pair selector
    const int row  = tile * 16 + r;
    const int lbl  = labels[row];

    const float* __restrict__ frow = feats   + (size_t)row * FEAT_DIM;
    const float* __restrict__ crow = centers + (size_t)lbl * FEAT_DIM;

    v8f acc = {};
    float sf = 0.0f, sc = 0.0f;
#pragma unroll
    for (int s = 0; s < FEAT_DIM / 4; ++s) {
        const int k = 4 * s + 2 * half;
        v2f a, b;
        a.x = frow[k];     a.y = frow[k + 1];
        b.x = crow[k];     b.y = crow[k + 1];
        sf += a.x * a.x + a.y * a.y;
        sc += b.x * b.x + b.y * b.y;
        // D += A x B   (V_WMMA_F32_16X16X4_F32, wave32, EXEC all-ones)
        acc = __builtin_amdgcn_wmma_f32_16x16x4_f32(
            false, a, false, b, (short)0, acc, false, false);
    }

    // full ||f_row||^2, ||c_row||^2 (lane r + lane r+16 hold complementary K halves)
    sf += __shfl_xor(sf, 16, 32);
    sc += __shfl_xor(sc, 16, 32);

    // diagonal dot product lives at c[lane&7] on lanes [0,8) and [24,32)
    const int ci = lane & 7;
    float dot = acc[0];
#pragma unroll
    for (int j = 1; j < 8; ++j) dot = (ci == j) ? acc[j] : dot;

    float contrib = 0.0f;
    if (lane < 8 || lane >= 24) {
        float d = sf + sc - 2.0f * dot;
        contrib = fminf(fmaxf(d, CLIP_MIN), CLIP_MAX);
    }

    // fixed-order wave reduction -> deterministic
#pragma unroll
    for (int off = 16; off > 0; off >>= 1)
        contrib += __shfl_xor(contrib, off, 32);

    if (lane == 0) partials[tile] = contrib;
}

// ---------------------------------------------------------------------------
// Kernel 3: deterministic serial finalize.
// loss = sum(partials)/N + (C-1)*1e-12   (the (N*C - N) clipped zeros)
// ---------------------------------------------------------------------------
__global__ void finalize_kernel(const float* __restrict__ partials,
                                float* __restrict__ out,
                                int ntiles, int N, int C) {
    if (threadIdx.x == 0 && blockIdx.x == 0) {
        float s = 0.0f;
        for (int i = 0; i < ntiles; ++i) s += partials[i];
        out[0] = s / (float)N + (float)(C - 1) * CLIP_MIN;
    }
}

extern "C" void kernel_launch(void* const* d_in, const int* in_sizes, int n_in,
                              void* d_out, int out_size, void* d_ws, size_t ws_size,
                              hipStream_t stream) {
    const float* logits  = (const float*)d_in[0];
    const float* feats   = (const float*)d_in[1];
    const float* centers = (const float*)d_in[2];

    const int N = in_sizes[1] / FEAT_DIM;     // 64*80 = 5120
    const int C = in_sizes[0] / N;            // 6625
    const int ntiles = N / 16;                // 320 (N divisible by 16)

    int*   labels   = (int*)d_ws;
    float* partials = (float*)((char*)d_ws + (size_t)N * sizeof(int));
    float* out      = (float*)d_out;

    // 1) argmax over logits (memory-roofline step: ~135 MB @ 23.3 TB/s)
    argmax_rows_kernel<<<N, 256, 0, stream>>>(logits, labels, C);

    // 2) WMMA distance tiles: 8 waves/block, 16 samples/wave
    center_dist_wmma_kernel<<<(ntiles + 7) / 8, 256, 0, stream>>>(
        feats, centers, labels, partials, ntiles);

    // 3) deterministic finalize
    finalize_kernel<<<1, 1, 0, stream>>>(partials, out, ntiles, N, C);
}